// GraphConv_2130303778989
// MI455X (gfx1250) — compile-verified
//
#include <hip/hip_runtime.h>

typedef __attribute__((ext_vector_type(2))) float v2f;
typedef __attribute__((ext_vector_type(8))) float v8f;

#define N_NODES 50000
#define N_EDGES 1600000
#define DIM     128
#define BATCH   2

// ---------------- kernel 1: zero degree buffer (ws is poisoned) --------------
__global__ void k_zero_deg(int* __restrict__ deg) {
    int i = blockIdx.x * 256 + threadIdx.x;
    if (i < N_NODES) deg[i] = 0;
}

// ---------------- kernel 2: degree count over src column ---------------------
__global__ void k_count_deg(const int* __restrict__ adj, int* __restrict__ deg) {
    int e = blockIdx.x * 256 + threadIdx.x;
    if (e < N_EDGES) atomicAdd(&deg[adj[2 * e]], 1);
}

// ---------------- kernel 3: node_w = rsqrt(deg + 1) --------------------------
__global__ void k_node_w(const int* __restrict__ deg, float* __restrict__ nw) {
    int i = blockIdx.x * 256 + threadIdx.x;
    if (i < N_NODES) nw[i] = rsqrtf((float)deg[i] + 1.0f);
}

// ---------------- kernel 4: x = inputs @ W via V_WMMA_F32_16X16X4_F32 --------
// Grid: (BATCH*N_NODES)/16 = 6250 blocks of 256 threads (8 waves).
// Each wave owns one 16x16 output tile (wave w -> columns w*16..w*16+15).
// A tile (16 rows x 128 K) is staged into LDS via the CDNA5 async pipe
// (GLOBAL_LOAD_ASYNC_TO_LDS_B128, tracked by ASYNCcnt): 2 B128 ops per thread
// copy 16 B/lane each directly to LDS without touching VGPRs. Destination row
// stride is padded to 132 floats (528 B: 16B-aligned rows, 132 mod 64 = 4 so
// the 16 rows hit distinct banks on the strided column reads).
// Fused epilogue: x -> workspace (for the edge gather), and
// out = node_w^2 * x + bias -> d_out (scatter kernel then atomically adds agg).
__global__ void __launch_bounds__(256, 2)
k_gemm_xw(const float* __restrict__ inp, const float* __restrict__ Wm,
          const float* __restrict__ bias, const float* __restrict__ nw,
          float* __restrict__ x, float* __restrict__ out) {
    __shared__ float lds_a[16][132];

    const int tid  = threadIdx.x;
    const int wave = tid >> 5;
    const int lane = tid & 31;
    const int row0 = blockIdx.x * 16;          // first of 16 rows (flattened b*N+node)

    // Async A-tile stage: thread t owns elements [8t, 8t+8) of the 16x128 tile.
    // INST_OFFSET is added to BOTH the LDS and global addresses (ISA 08 §4.4),
    // so offset:16 covers the second float4 of each thread's 32-byte slice.
    {
        int r = tid >> 4;                      // 0..15 (dest row)
        int c = (tid & 15) * 8;                // 0,8,...,120 (dest col)
        unsigned lds0 = (unsigned)(size_t)&lds_a[r][c];          // LDS byte addr
        unsigned voff = (unsigned)(tid * 8u * sizeof(float));    // global byte off
        const float* gbase = inp + (size_t)row0 * DIM;           // uniform -> SGPR
        asm volatile(
            "global_load_async_to_lds_b128 %0, %1, %2\n\t"
            "global_load_async_to_lds_b128 %0, %1, %2 offset:16"
            :
            : "v"(lds0), "v"(voff), "s"(gbase)
            : "memory");
        asm volatile("s_wait_asynccnt 0x0" ::: "memory");        // this wave's copies
    }
    __syncthreads();                                             // cross-wave visibility

    const int n0   = wave * 16;                // output-column base for this wave
    const int mrow = lane & 15;                // A row / B column within tile
    const int kh   = lane >> 4;                // lanes 16-31 carry K+2 (per ISA table)

    v8f acc = {};
    #pragma unroll
    for (int k = 0; k < DIM; k += 4) {
        v2f a, b;
        a.x = lds_a[mrow][k + 2 * kh + 0];
        a.y = lds_a[mrow][k + 2 * kh + 1];
        b.x = Wm[(k + 2 * kh + 0) * DIM + n0 + mrow];
        b.y = Wm[(k + 2 * kh + 1) * DIM + n0 + mrow];
        acc = __builtin_amdgcn_wmma_f32_16x16x4_f32(
                  /*neg_a=*/false, a, /*neg_b=*/false, b,
                  /*c_mod=*/(short)0, acc, /*reuse_a=*/false, /*reuse_b=*/false);
    }

    // Epilogue. C/D layout: VGPR i -> M=i (lanes 0-15) / M=i+8 (lanes 16-31).
    const float bv = bias[n0 + mrow];
    #pragma unroll
    for (int i = 0; i < 8; i++) {
        int   m    = i + (kh << 3);
        int   row  = row0 + m;                 // 50000 % 16 == 0: no batch straddle
        int   node = row % N_NODES;
        size_t off = (size_t)row * DIM + n0 + mrow;
        float xv   = acc[i];
        x[off] = xv;
        float w = nw[node];
        out[off] = w * w * xv + bv;
    }
}

// ---------------- kernel 5: edge scatter-add ---------------------------------
// One wave per (edge, batch): float4 gather of the 128-float source row from
// x (L2-resident, 51.2 MB < 192 MB L2), scale by edge_w, 4 f32 atomics/lane.
__global__ void __launch_bounds__(256)
k_scatter(const int* __restrict__ adj, const float* __restrict__ nw,
          const float* __restrict__ x, float* __restrict__ out) {
    long t    = (long)blockIdx.x * 256 + threadIdx.x;
    long pair = t >> 5;
    int  lane = threadIdx.x & 31;
    if (pair >= (long)N_EDGES * BATCH) return;

    int e = (int)(pair >> 1);
    int b = (int)(pair & 1);
    int s = adj[2 * e + 0];
    int d = adj[2 * e + 1];
    float ew = nw[s] * nw[d];

    const float4* xs = (const float4*)(x + ((size_t)b * N_NODES + s) * DIM);
    float*        op = out + ((size_t)b * N_NODES + d) * DIM;

    float4 v = xs[lane];                       // 32 lanes x 4 = 128 floats
    atomicAdd(&op[lane * 4 + 0], v.x * ew);
    atomicAdd(&op[lane * 4 + 1], v.y * ew);
    atomicAdd(&op[lane * 4 + 2], v.z * ew);
    atomicAdd(&op[lane * 4 + 3], v.w * ew);
}

// ---------------- host-side launch -------------------------------------------
extern "C" void kernel_launch(void* const* d_in, const int* in_sizes, int n_in,
                              void* d_out, int out_size, void* d_ws, size_t ws_size,
                              hipStream_t stream) {
    const float* inp  = (const float*)d_in[0];   // (2, 50000, 128) f32
    const float* Wm   = (const float*)d_in[1];   // (128, 128) f32
    const float* bias = (const float*)d_in[2];   // (1, 1, 128) f32
    const int*   adj  = (const int*)d_in[3];     // (1.6M, 2) i32
    float*       out  = (float*)d_out;           // (2, 50000, 128) f32

    // Workspace layout: deg[int x 50000] | nw[float x 50000] | x[float x 12.8M]
    // x offset = 400000 B (16B-aligned for float4 access). Total ~51.6 MB.
    char*  ws  = (char*)d_ws;
    int*   deg = (int*)ws;
    float* nw  = (float*)(ws + (size_t)N_NODES * sizeof(int));
    float* x   = (float*)(ws + (size_t)2 * N_NODES * sizeof(int));

    k_zero_deg <<<(N_NODES + 255) / 256, 256, 0, stream>>>(deg);
    k_count_deg<<<(N_EDGES + 255) / 256, 256, 0, stream>>>(adj, deg);
    k_node_w   <<<(N_NODES + 255) / 256, 256, 0, stream>>>(deg, nw);
    k_gemm_xw  <<<(BATCH * N_NODES) / 16, 256, 0, stream>>>(inp, Wm, bias, nw, x, out);

    long scatter_threads = (long)N_EDGES * BATCH * 32;
    k_scatter  <<<(int)((scatter_threads + 255) / 256), 256, 0, stream>>>(adj, nw, x, out);
}